// SpatialBeamAttention_15350213116295
// MI455X (gfx1250) — compile-verified
//
#include <hip/hip_runtime.h>
#include <hip/hip_bf16.h>

// Problem constants (fixed by the reference setup_inputs)
#define BB   2
#define CCT  128
#define CB   32
#define NN   179
#define DD   24
#define VV   (DD*DD*DD)        // 13824
#define VTILES_PER_B (VV/16)   // 864

typedef float v2f __attribute__((ext_vector_type(2)));
typedef float v8f __attribute__((ext_vector_type(8)));

// ---------------------------------------------------------------------------
// Kernel 1: q[b,v,o] = sum_c wq[o,c] * ct[b,c,v] + bq[o]
// Per-wave 16(v) x 32(o) tile via V_WMMA_F32_16X16X4_F32, K looped in steps of 4.
// A fragment (16x4 f32): lane<16 holds M=lane, VGPR0=K0, VGPR1=K1;
//                        lane>=16 holds M=lane-16, VGPR0=K2, VGPR1=K3.
// B fragment (4x16 f32): lane<16 N=lane {K0,K1}; lane>=16 N=lane-16 {K2,K3}.
// D (16x16 f32): VGPR r: lanes0-15 -> [M=r][N=lane]; lanes16-31 -> [M=8+r][N=lane-16]
// ct is read once (14 MB): non-temporal so it doesn't claim L2.
// ---------------------------------------------------------------------------
__global__ void __launch_bounds__(128)
qproj_wmma_kernel(const float* __restrict__ ct, const float* __restrict__ wq,
                  const float* __restrict__ bq, float* __restrict__ qout) {
    const int wave = threadIdx.x >> 5;
    const int lane = threadIdx.x & 31;
    const int half = lane >> 4;     // 0 | 1
    const int mrow = lane & 15;

    const int tileId = blockIdx.x * 4 + wave;       // 0 .. B*864-1
    const int b  = tileId / VTILES_PER_B;
    const int v0 = (tileId - b * VTILES_PER_B) * 16;

    const float* ctb = ct + (size_t)b * CCT * VV;

    v8f acc0 = {};   // o-tile [0,16)
    v8f acc1 = {};   // o-tile [16,32)

    #pragma unroll 4
    for (int k0 = 0; k0 < CCT; k0 += 4) {
        const int c = k0 + 2 * half;
        v2f a;
        a.x = __builtin_nontemporal_load(&ctb[(size_t)c       * VV + v0 + mrow]);
        a.y = __builtin_nontemporal_load(&ctb[(size_t)(c + 1) * VV + v0 + mrow]);
        v2f bf0 = *(const v2f*)(wq + (size_t)(mrow)      * CCT + c);
        v2f bf1 = *(const v2f*)(wq + (size_t)(16 + mrow) * CCT + c);
        acc0 = __builtin_amdgcn_wmma_f32_16x16x4_f32(false, a, false, bf0,
                                                     (short)0, acc0, false, false);
        acc1 = __builtin_amdgcn_wmma_f32_16x16x4_f32(false, a, false, bf1,
                                                     (short)0, acc1, false, false);
    }

    const float bias0 = bq[mrow];
    const float bias1 = bq[16 + mrow];
    const int   n     = mrow;
    const int   mbase = half * 8;
    float* qb = qout + ((size_t)b * VV + v0) * CB;

    #pragma unroll
    for (int r = 0; r < 8; ++r) {
        const int m = mbase + r;
        qb[(size_t)m * CB + n]      = acc0[r] + bias0;
        qb[(size_t)m * CB + 16 + n] = acc1[r] + bias1;
    }
}

// ---------------------------------------------------------------------------
// Kernel 2: single streaming pass over beam_stack (633 MB, non-temporal:
// stream is 3.3x the 192 MB L2, so NT keeps L2 for the logit slab + q buffer).
// thread = voxel. Online softmax; writes raw logits to attn region of d_out,
// per-voxel (m, 1/l) to workspace, and LayerNormed fused output.
// ---------------------------------------------------------------------------
__global__ void __launch_bounds__(256)
attn_fused_kernel(const float* __restrict__ q, const float* __restrict__ beam,
                  const float* __restrict__ gamma, const float* __restrict__ beta,
                  float* __restrict__ fused_out, float* __restrict__ attn_logits,
                  float* __restrict__ m_out, float* __restrict__ linv_out) {
    const int tid = blockIdx.x * blockDim.x + threadIdx.x;   // 0 .. B*V-1
    const int b = tid / VV;
    const int v = tid - b * VV;
    const float scale = 0.1767766952966369f;                 // 32^-0.5

    // q[32] for this voxel (contiguous, float4 loads)
    float qr[CB];
    const float4* qv = (const float4*)(q + (size_t)tid * CB);
    #pragma unroll
    for (int i = 0; i < CB / 4; ++i) {
        float4 t = qv[i];
        qr[4*i+0] = t.x; qr[4*i+1] = t.y; qr[4*i+2] = t.z; qr[4*i+3] = t.w;
    }

    float acc[CB];
    #pragma unroll
    for (int c = 0; c < CB; ++c) acc[c] = 0.0f;
    float m = -3.0e38f;
    float l = 0.0f;

    const float* bb = beam + (size_t)b * NN * CB * VV + v;
    float* lg = attn_logits + (size_t)tid * NN;

    for (int n = 0; n < NN; ++n) {
        const float* p = bb + (size_t)n * CB * VV;
        float kv[CB];
        #pragma unroll
        for (int c = 0; c < CB; ++c)
            kv[c] = __builtin_nontemporal_load(&p[(size_t)c * VV]); // coalesced in v
        float dot = 0.0f;
        #pragma unroll
        for (int c = 0; c < CB; ++c) dot = __builtin_fmaf(qr[c], kv[c], dot);
        const float logit = dot * scale;
        lg[n] = logit;                                          // raw logit out (RT: L2 write-combine)

        if (logit > m) {                 // rare path: new running max
            const float f = __expf(m - logit);                  // 0 on first iter
            l = l * f + 1.0f;
            #pragma unroll
            for (int c = 0; c < CB; ++c) acc[c] = acc[c] * f + kv[c];
            m = logit;
        } else {                          // common path
            const float pe = __expf(logit - m);
            l += pe;
            #pragma unroll
            for (int c = 0; c < CB; ++c) acc[c] = __builtin_fmaf(pe, kv[c], acc[c]);
        }
    }

    const float linv = 1.0f / l;

    // LayerNorm over the 32 channels
    float mu = 0.0f;
    #pragma unroll
    for (int c = 0; c < CB; ++c) { acc[c] *= linv; mu += acc[c]; }
    mu *= (1.0f / CB);
    float var = 0.0f;
    #pragma unroll
    for (int c = 0; c < CB; ++c) { const float d = acc[c] - mu; var = __builtin_fmaf(d, d, var); }
    var *= (1.0f / CB);
    const float rstd = rsqrtf(var + 1e-5f);

    float* fo = fused_out + (size_t)b * CB * VV + v;
    #pragma unroll
    for (int c = 0; c < CB; ++c)
        fo[(size_t)c * VV] = (acc[c] - mu) * rstd * gamma[c] + beta[c]; // coalesced

    m_out[tid]    = m;
    linv_out[tid] = linv;
}

// ---------------------------------------------------------------------------
// Kernel 3: attn_w = exp(logit - m) / l   (in place on d_out attn region)
// ---------------------------------------------------------------------------
__global__ void __launch_bounds__(256)
attn_finalize_kernel(float* __restrict__ attn, const float* __restrict__ m,
                     const float* __restrict__ linv, int total) {
    const int idx = blockIdx.x * blockDim.x + threadIdx.x;
    if (idx >= total) return;
    const int bv = idx / NN;
    attn[idx] = __expf(attn[idx] - m[bv]) * linv[bv];
}

extern "C" void kernel_launch(void* const* d_in, const int* in_sizes, int n_in,
                              void* d_out, int out_size, void* d_ws, size_t ws_size,
                              hipStream_t stream) {
    const float* ct    = (const float*)d_in[0];   // [B,128,24,24,24]
    const float* beam  = (const float*)d_in[1];   // [B,179,32,24,24,24]
    const float* wq    = (const float*)d_in[2];   // [32,128]
    const float* bq    = (const float*)d_in[3];   // [32]
    const float* gamma = (const float*)d_in[4];   // [32]
    const float* beta  = (const float*)d_in[5];   // [32]

    float* fused_out = (float*)d_out;                         // B*32*V = 884736
    float* attn_out  = fused_out + (size_t)BB * CB * VV;      // B*V*N  = 4948992

    float* q_ws    = (float*)d_ws;                            // B*V*32 floats
    float* m_ws    = q_ws + (size_t)BB * VV * CB;             // B*V floats
    float* linv_ws = m_ws + (size_t)BB * VV;                  // B*V floats

    // 1) q projection: B*864 v-tiles, 4 waves (tiles) per 128-thread block
    qproj_wmma_kernel<<<(BB * VTILES_PER_B) / 4, 128, 0, stream>>>(ct, wq, bq, q_ws);

    // 2) fused attention pass: B*V threads
    attn_fused_kernel<<<(BB * VV) / 256, 256, 0, stream>>>(
        q_ws, beam, gamma, beta, fused_out, attn_out, m_ws, linv_ws);

    // 3) softmax finalize on attn weights
    const int total = BB * VV * NN;
    attn_finalize_kernel<<<(total + 255) / 256, 256, 0, stream>>>(
        attn_out, m_ws, linv_ws, total);
}